// GINEncoder_44504451121830
// MI455X (gfx1250) — compile-verified
//
#include <hip/hip_runtime.h>

#define NNODES 50000
#define NEDGES 800000
#define HID 128
#define NGRAPHS 128
#define SXS 130  // padded LDS stride for r1 tile: bank = 2*row+k (conflict-free), 8B aligned

typedef __attribute__((ext_vector_type(2))) float v2f;
typedef __attribute__((ext_vector_type(8))) float v8f;
typedef __attribute__((ext_vector_type(4))) unsigned int u32x4;
typedef __attribute__((ext_vector_type(4))) int i32x4;
typedef __attribute__((ext_vector_type(8))) int i32x8;

#if defined(__has_builtin)
#if __has_builtin(__builtin_amdgcn_tensor_load_to_lds)
#define GIN_HAVE_TDM 1
#endif
#endif

// ---------------------------------------------------------------- z = (1+eps)*h
__global__ void scale_init_kernel(float* __restrict__ z, const float* __restrict__ h,
                                  const float* __restrict__ eps, int layer, int n4) {
  int i = blockIdx.x * blockDim.x + threadIdx.x;
  if (i >= n4) return;
  float s = 1.0f + eps[layer];
  float4 v = ((const float4*)h)[i];
  v.x *= s; v.y *= s; v.z *= s; v.w *= s;
  ((float4*)z)[i] = v;
}

// ------------------------------------------------- scatter-add: z[dst] += h[src]
__global__ void edge_scatter_kernel(const float* __restrict__ h, const int* __restrict__ src,
                                    const int* __restrict__ dst, float* __restrict__ z, int ne) {
  int e = blockIdx.x * (blockDim.x >> 5) + (threadIdx.x >> 5);
  if (e >= ne) return;
  int lane = threadIdx.x & 31;
  long long s = src[e], d = dst[e];
  float4 v = ((const float4*)(h + s * HID))[lane];
  float* zp = z + d * HID + lane * 4;
  unsafeAtomicAdd(zp + 0, v.x);
  unsafeAtomicAdd(zp + 1, v.y);
  unsafeAtomicAdd(zp + 2, v.z);
  unsafeAtomicAdd(zp + 3, v.w);
}

#ifdef GIN_HAVE_TDM
// TDM copy of one 128x128 f32 weight matrix (64 KB, contiguous) into LDS.
// D# built per CDNA5 ISA: 1D tile of 16384 4-byte elems.
__device__ __forceinline__ void tdm_copy_w_to_lds(const float* __restrict__ gsrc, float* ldsdst) {
  const unsigned lds_off = (unsigned)(unsigned long long)(void*)ldsdst;  // LDS aperture: addr[31:0]
  const unsigned long long ga = (unsigned long long)gsrc;
  u32x4 g0;
  g0.x = 1u;                                  // count=1, user descriptor
  g0.y = lds_off;                             // lds_addr (bits 63:32)
  g0.z = (unsigned)ga;                        // global_addr[31:0]
  g0.w = (unsigned)(ga >> 32) | 0x80000000u;  // global_addr[56:32] | type=2 (bits127:126)
  i32x8 g1 = {};
  g1[0] = 0x00020000;       // workgroup_mask=0, data_size=2 (4 bytes)
  g1[1] = (int)0x40000000;  // tensor_dim0 = 16384  (bits 79:48 -> word1[31:16])
  g1[2] = 0x00010000;       // tensor_dim1 = 1      (bits 111:80 -> word2[31:16])
  g1[3] = (int)0x40000000;  // tile_dim0 = 16384    (bits 127:112)
  g1[4] = 1;                // tile_dim1 = 1
  g1[5] = 16384;            // tensor_dim0_stride = 16384
  i32x4 gz = {};
#if __clang_major__ >= 23
  i32x8 gz8 = {};
  __builtin_amdgcn_tensor_load_to_lds(g0, g1, gz, gz, gz8, 0);
#else
  __builtin_amdgcn_tensor_load_to_lds(g0, g1, gz, gz, 0);
#endif
}

__device__ __forceinline__ void tdm_wait() {
#if __has_builtin(__builtin_amdgcn_s_wait_tensorcnt)
  __builtin_amdgcn_s_wait_tensorcnt(0);
#else
  asm volatile("s_wait_tensorcnt 0x0" ::: "memory");
#endif
}
#endif

// --------------------------------------- fused MLP: hout = relu(z@W1+b1)@W2+b2
// 256 threads = 8 waves; wave owns 16 rows; 128 rows/block.
// f32 WMMA 16x16x4: A 16x4 (lane&15 = M, (lane>>4)*2+j = K),
//                   B 4x16 ((lane>>4)*2+j = K, lane&15 = N),
//                   C/D 16x16 (vgpr r + (lane>>4)*8 = M, lane&15 = N).
__global__ __launch_bounds__(256, 2)
void gin_mlp_kernel(const float* __restrict__ z, const float* __restrict__ W1,
                    const float* __restrict__ b1, const float* __restrict__ W2,
                    const float* __restrict__ b2, float* __restrict__ hout, int n) {
  __shared__ float sW[HID * HID];   // 64 KB: W1, then W2 after barrier
  __shared__ float sX[HID * SXS];   // 65 KB: relu intermediate (padded stride)
  __shared__ float sb[HID];

  const int tid  = threadIdx.x;
  const int lane = tid & 31;
  const int wv   = tid >> 5;
  const int half = lane >> 4;
  const int l16  = lane & 15;
  const int rowbase = blockIdx.x * 128 + wv * 16;

  // ---- stage W1 + b1 into LDS
#ifdef GIN_HAVE_TDM
  if (wv == 0) {
    tdm_copy_w_to_lds(W1, sW);
    tdm_wait();
  }
  if (tid < HID) sb[tid] = b1[tid];
#else
  {
    const float4* g = (const float4*)W1;
    float4* s4 = (float4*)sW;
#pragma unroll
    for (int i = 0; i < (HID * HID / 4) / 256; ++i) s4[tid + i * 256] = g[tid + i * 256];
    if (tid < HID) sb[tid] = b1[tid];
  }
#endif
  __syncthreads();

  // A fragments for this wave's 16 rows of z (zero past n; no divergence at WMMA)
  v2f afrag[32];
  {
    const int row = rowbase + l16;
    const v2f* zr = (const v2f*)(z + (long long)row * HID + half * 2);  // 8B aligned
#pragma unroll
    for (int ks = 0; ks < 32; ++ks) {
      v2f a; a.x = 0.f; a.y = 0.f;
      if (row < n) a = zr[ks * 2];
      afrag[ks] = a;
    }
  }

  // ---- stage 1: r1 = relu(z @ W1 + b1) -> LDS (C-layout order, stride SXS)
#pragma unroll
  for (int nt = 0; nt < 8; ++nt) {
    v8f acc = {};
    const int coln = nt * 16 + l16;
    const float* wcol = sW + (half * 2) * HID + coln;  // constant k-offsets -> ds_load_2addr_st64
#pragma unroll
    for (int ks = 0; ks < 32; ++ks) {
      v2f b;
      b.x = wcol[ks * 4 * HID];
      b.y = wcol[ks * 4 * HID + HID];
      acc = __builtin_amdgcn_wmma_f32_16x16x4_f32(false, afrag[ks], false, b,
                                                  (short)0, acc, false, false);
    }
    const float bias = sb[coln];
#pragma unroll
    for (int r = 0; r < 8; ++r) {
      float v = acc[r] + bias;
      sX[(wv * 16 + r + half * 8) * SXS + coln] = v > 0.f ? v : 0.f;
    }
  }

  // re-read own rows in A-fragment layout (wave-private region; DS in-order per wave)
  {
    const v2f* xr = (const v2f*)(sX + (wv * 16 + l16) * SXS + half * 2);  // 8B aligned
#pragma unroll
    for (int ks = 0; ks < 32; ++ks) afrag[ks] = xr[ks * 2];
  }
  __syncthreads();  // all waves done reading W1 from sW

  // ---- stage W2 + b2 into LDS
#ifdef GIN_HAVE_TDM
  if (wv == 0) {
    tdm_copy_w_to_lds(W2, sW);
    tdm_wait();
  }
  if (tid < HID) sb[tid] = b2[tid];
#else
  {
    const float4* g = (const float4*)W2;
    float4* s4 = (float4*)sW;
#pragma unroll
    for (int i = 0; i < (HID * HID / 4) / 256; ++i) s4[tid + i * 256] = g[tid + i * 256];
    if (tid < HID) sb[tid] = b2[tid];
  }
#endif
  __syncthreads();

  // ---- stage 2: h = r1 @ W2 + b2
#pragma unroll
  for (int nt = 0; nt < 8; ++nt) {
    v8f acc = {};
    const int coln = nt * 16 + l16;
    const float* wcol = sW + (half * 2) * HID + coln;
#pragma unroll
    for (int ks = 0; ks < 32; ++ks) {
      v2f b;
      b.x = wcol[ks * 4 * HID];
      b.y = wcol[ks * 4 * HID + HID];
      acc = __builtin_amdgcn_wmma_f32_16x16x4_f32(false, afrag[ks], false, b,
                                                  (short)0, acc, false, false);
    }
    const float bias = sb[coln];
#pragma unroll
    for (int r = 0; r < 8; ++r) {
      const int grow = rowbase + r + half * 8;
      if (grow < n) hout[(long long)grow * HID + coln] = acc[r] + bias;
    }
  }
}

// ------------------------------------------------------------- graph sum pooling
__global__ void zero_kernel(float* p, int n) {
  int i = blockIdx.x * blockDim.x + threadIdx.x;
  if (i < n) p[i] = 0.f;
}

__global__ void pool_kernel(const float* __restrict__ h, const int* __restrict__ gid,
                            float* __restrict__ out, int n) {
  int node = blockIdx.x * (blockDim.x >> 5) + (threadIdx.x >> 5);
  if (node >= n) return;
  int lane = threadIdx.x & 31;
  int g = gid[node];
  float4 v = ((const float4*)(h + (long long)node * HID))[lane];
  float* op = out + (long long)g * HID + lane * 4;
  unsafeAtomicAdd(op + 0, v.x);
  unsafeAtomicAdd(op + 1, v.y);
  unsafeAtomicAdd(op + 2, v.z);
  unsafeAtomicAdd(op + 3, v.w);
}

// --------------------------------------------------------------------- launcher
extern "C" void kernel_launch(void* const* d_in, const int* in_sizes, int n_in,
                              void* d_out, int out_size, void* d_ws, size_t ws_size,
                              hipStream_t stream) {
  const float* feats = (const float*)d_in[0];
  const int*   esrc  = (const int*)d_in[1];
  const int*   edst  = (const int*)d_in[2];
  const int*   gid   = (const int*)d_in[3];
  const float* eps   = (const float*)d_in[4];
  const float* W1[3] = {(const float*)d_in[5],  (const float*)d_in[9],  (const float*)d_in[13]};
  const float* B1[3] = {(const float*)d_in[6],  (const float*)d_in[10], (const float*)d_in[14]};
  const float* W2[3] = {(const float*)d_in[7],  (const float*)d_in[11], (const float*)d_in[15]};
  const float* B2[3] = {(const float*)d_in[8],  (const float*)d_in[12], (const float*)d_in[16]};
  float* out = (float*)d_out;

  float* h_ws = (float*)d_ws;                 // [NNODES, HID]
  float* z    = h_ws + (size_t)NNODES * HID;  // [NNODES, HID]

  const int n4 = NNODES * HID / 4;
  for (int layer = 0; layer < 3; ++layer) {
    const float* hin = (layer == 0) ? feats : h_ws;
    scale_init_kernel<<<(n4 + 255) / 256, 256, 0, stream>>>(z, hin, eps, layer, n4);
    edge_scatter_kernel<<<(NEDGES + 7) / 8, 256, 0, stream>>>(hin, esrc, edst, z, NEDGES);
    gin_mlp_kernel<<<(NNODES + 127) / 128, 256, 0, stream>>>(
        z, W1[layer], B1[layer], W2[layer], B2[layer], h_ws, NNODES);
  }
  zero_kernel<<<(NGRAPHS * HID + 255) / 256, 256, 0, stream>>>(out, NGRAPHS * HID);
  pool_kernel<<<(NNODES + 7) / 8, 256, 0, stream>>>(h_ws, gid, out, NNODES);
}